// Piston_PolicyHelperCASE_51445118271681
// MI455X (gfx1250) — compile-verified
//
#include <hip/hip_runtime.h>
#include <hip/hip_bf16.h>

typedef float v2f __attribute__((ext_vector_type(2)));
typedef float v8f __attribute__((ext_vector_type(8)));

#define WMMA_F32X4(A, Bm, C) \
    __builtin_amdgcn_wmma_f32_16x16x4_f32(false, (A), false, (Bm), (short)0, (C), false, false)

constexpr int Bsz = 262144;   // batch
constexpr int Kn  = 8;        // neighbors (sequential steps)
constexpr int Hh  = 10;       // hidden
constexpr int G3  = 30;       // 3*H gate width
constexpr int HP  = 12;       // padded h row stride (zeros in cols 10,11)
constexpr int GW  = 32;       // gate tile row stride in LDS

// per-wave LDS floats: h(16*12=192) + gi(16*32=512) + gh(16*32=512)
constexpr int WAVE_LDS = 192 + 512 + 512;

__device__ __forceinline__ float sigf(float x) {
    float u = __builtin_amdgcn_exp2f(-1.4426950408889634f * x);
    return __builtin_amdgcn_rcpf(1.0f + u);
}
__device__ __forceinline__ float tanh_fast(float x) {
    x = fminf(fmaxf(x, -10.0f), 10.0f);
    float t = __builtin_amdgcn_exp2f(2.8853900817779268f * x); // exp(2x)
    return (t - 1.0f) * __builtin_amdgcn_rcpf(t + 1.0f);
}

__global__ __launch_bounds__(256)
void Piston_GRUComm_wmma_kernel(const float* __restrict__ policy,
                                const float* __restrict__ neighbors,
                                const float* __restrict__ w_ih,
                                const float* __restrict__ w_hh,
                                const float* __restrict__ b_ih,
                                const float* __restrict__ b_hh,
                                float* __restrict__ out)
{
    __shared__ float lds_all[8 * WAVE_LDS];
    const int lane = threadIdx.x & 31;
    const int wave = threadIdx.x >> 5;
    float* hbuf  = lds_all + wave * WAVE_LDS;
    float* gibuf = hbuf + 192;
    float* ghbuf = gibuf + 512;

    const int r0   = (blockIdx.x * 8 + wave) * 16;  // 16 batch rows per wave
    const int mrow = lane & 15;
    const int hi   = lane >> 4;                      // half-wave id (A/B layout)

    // ---- weights once, in B-layout fragments (K-chunk c covers K=4c..4c+3) ----
    v2f wih[2][3], whh[2][3];
    #pragma unroll
    for (int t = 0; t < 2; ++t) {
        #pragma unroll
        for (int c = 0; c < 3; ++c) {
            int rw = t * 16 + mrow;        // gate/output column of W (N dim)
            int c0 = 4 * c + 2 * hi;       // K index base for this half-wave
            float a0 = (rw < G3 && c0     < Hh) ? w_ih[rw * Hh + c0]     : 0.0f;
            float a1 = (rw < G3 && c0 + 1 < Hh) ? w_ih[rw * Hh + c0 + 1] : 0.0f;
            float b0 = (rw < G3 && c0     < Hh) ? w_hh[rw * Hh + c0]     : 0.0f;
            float b1 = (rw < G3 && c0 + 1 < Hh) ? w_hh[rw * Hh + c0 + 1] : 0.0f;
            wih[t][c] = (v2f){a0, a1};
            whh[t][c] = (v2f){b0, b1};
        }
    }

    // ---- biases as column-broadcast C initializers ----
    const int n1 = 16 + mrow;
    const float bi0 = b_ih[mrow];
    const float bh0 = b_hh[mrow];
    const float bi1 = (n1 < G3) ? b_ih[n1] : 0.0f;
    const float bh1 = (n1 < G3) ? b_hh[n1] : 0.0f;

    float* tempOut = out + (size_t)Bsz * Hh;   // [B, K+1, H] region

    // ---- h0 tile -> LDS, and temp_out[:,0,:] = policy ----
    #pragma unroll
    for (int e = 0; e < 5; ++e) {
        int idx = e * 32 + lane;           // 0..159 over 16 rows x 10 cols
        int m = idx / Hh, j = idx % Hh;
        float v = policy[(size_t)(r0 + m) * Hh + j];
        hbuf[m * HP + j] = v;
        tempOut[(size_t)(r0 + m) * (Kn + 1) * Hh + j] = v;
    }
    hbuf[mrow * HP + 10 + hi] = 0.0f;      // zero pad cols 10,11 (A-layout chunk 2)

    for (int k = 0; k < Kn; ++k) {
        __syncthreads();                   // h tile stable before A-layout reads

        // ---- A-layout fragments: x from global, h from LDS ----
        v2f xa[3], ha[3];
        const size_t xoff = ((size_t)(r0 + mrow) * Kn + (size_t)k) * Hh;
        #pragma unroll
        for (int c = 0; c < 3; ++c) {
            int c0 = 4 * c + 2 * hi;
            float x0 = (c0     < Hh) ? neighbors[xoff + c0]     : 0.0f;
            float x1 = (c0 + 1 < Hh) ? neighbors[xoff + c0 + 1] : 0.0f;
            xa[c] = (v2f){x0, x1};
            ha[c] = (v2f){hbuf[mrow * HP + c0], hbuf[mrow * HP + c0 + 1]};
        }

        // ---- 12x v_wmma_f32_16x16x4_f32: gi and gh tile pairs ----
        v8f aI0, aI1, aH0, aH1;
        #pragma unroll
        for (int i = 0; i < 8; ++i) { aI0[i] = bi0; aI1[i] = bi1; aH0[i] = bh0; aH1[i] = bh1; }
        #pragma unroll
        for (int c = 0; c < 3; ++c) {
            aI0 = WMMA_F32X4(xa[c], wih[0][c], aI0);
            aI1 = WMMA_F32X4(xa[c], wih[1][c], aI1);
            aH0 = WMMA_F32X4(ha[c], whh[0][c], aH0);
            aH1 = WMMA_F32X4(ha[c], whh[1][c], aH1);
        }

        // ---- spill gate pre-activations (C-layout) to LDS ----
        #pragma unroll
        for (int v = 0; v < 8; ++v) {
            int m = v + 8 * hi;
            gibuf[m * GW + mrow]      = aI0[v];
            gibuf[m * GW + 16 + mrow] = aI1[v];
            ghbuf[m * GW + mrow]      = aH0[v];
            ghbuf[m * GW + 16 + mrow] = aH1[v];
        }
        __syncthreads();

        // ---- gate math: each lane owns 5 (row, j) elements ----
        #pragma unroll
        for (int e = 0; e < 5; ++e) {
            int idx = e * 32 + lane;
            int m = idx / Hh, j = idx % Hh;
            float gir = gibuf[m * GW + j],      ghr = ghbuf[m * GW + j];
            float giz = gibuf[m * GW + j + 10], ghz = ghbuf[m * GW + j + 10];
            float gin = gibuf[m * GW + j + 20], ghn = ghbuf[m * GW + j + 20];
            float r = sigf(gir + ghr);
            float z = sigf(giz + ghz);
            float n = tanh_fast(gin + r * ghn);
            float hold = hbuf[m * HP + j];
            float hnew = (1.0f - z) * n + z * hold;
            hbuf[m * HP + j] = hnew;
            tempOut[((size_t)(r0 + m) * (Kn + 1) + (size_t)(k + 1)) * Hh + j] = hnew;
            if (k == Kn - 1)
                out[(size_t)(r0 + m) * Hh + j] = hnew;   // h_final region
        }
    }
}

extern "C" void kernel_launch(void* const* d_in, const int* in_sizes, int n_in,
                              void* d_out, int out_size, void* d_ws, size_t ws_size,
                              hipStream_t stream) {
    (void)in_sizes; (void)n_in; (void)d_ws; (void)ws_size; (void)out_size;
    const float* policy    = (const float*)d_in[0];
    const float* neighbors = (const float*)d_in[1];
    const float* w_ih      = (const float*)d_in[2];
    const float* w_hh      = (const float*)d_in[3];
    const float* b_ih      = (const float*)d_in[4];
    const float* b_hh      = (const float*)d_in[5];
    float* out = (float*)d_out;

    const int rows_per_block = 16 * 8;                 // 16 rows/wave, 8 waves
    const int grid = Bsz / rows_per_block;             // 2048
    Piston_GRUComm_wmma_kernel<<<grid, 256, 0, stream>>>(
        policy, neighbors, w_ih, w_hh, b_ih, b_hh, out);
}